// WePredConv_83330955477483
// MI455X (gfx1250) — compile-verified
//
#include <hip/hip_runtime.h>
#include <math.h>

// Problem constants (from the reference)
#define N_NODES 100000
#define N_EDGES 1600000
#define DIM_IN  128
#define HEADS   4
#define CHAN    32
#define HC      128   // HEADS*CHAN
#define EDIM    64

typedef __attribute__((ext_vector_type(2))) float v2f;
typedef __attribute__((ext_vector_type(8))) float v8f;

// ---------------------------------------------------------------------------
// Float atomic max via monotone bit mapping (int max for >=0, uint min for <0).
// Correct for IEEE-754 when the cell is initialized to -inf (0xFF800000).
// Compiles to global_atomic_max_i32 / global_atomic_min_u32 (L2 atomics).
// ---------------------------------------------------------------------------
__device__ __forceinline__ void atomicMaxF(float* addr, float val) {
    if (val >= 0.0f) {
        atomicMax((int*)addr, __float_as_int(val));
    } else {
        atomicMin((unsigned int*)addr, __float_as_uint(val));
    }
}

// ---------------------------------------------------------------------------
// Kernel 1: collapse Wq/Wk/We over the C dimension.
//   Wq_s[k*4+h] = sum_c Wq[k, h*32+c]   (and same for Wk, We)
// ---------------------------------------------------------------------------
__global__ void k_reduce_w(const float* __restrict__ Wq,
                           const float* __restrict__ Wk,
                           const float* __restrict__ We,
                           float* __restrict__ Wq_s,
                           float* __restrict__ Wk_s,
                           float* __restrict__ We_s) {
    int i = blockIdx.x * blockDim.x + threadIdx.x;
    if (i < 512) {                       // Wq: 128 rows x 4 heads
        int k = i >> 2, h = i & 3;
        const float* p = Wq + k * HC + h * CHAN;
        float s = 0.f;
        #pragma unroll
        for (int c = 0; c < CHAN; ++c) s += p[c];
        Wq_s[i] = s;
    } else if (i < 1024) {               // Wk
        int j = i - 512;
        int k = j >> 2, h = j & 3;
        const float* p = Wk + k * HC + h * CHAN;
        float s = 0.f;
        #pragma unroll
        for (int c = 0; c < CHAN; ++c) s += p[c];
        Wk_s[j] = s;
    } else if (i < 1280) {               // We: 64 rows x 4 heads
        int j = i - 1024;
        int k = j >> 2, h = j & 3;
        const float* p = We + k * HC + h * CHAN;
        float s = 0.f;
        #pragma unroll
        for (int c = 0; c < CHAN; ++c) s += p[c];
        We_s[j] = s;
    }
}

// ---------------------------------------------------------------------------
// Kernel 2: init out = bias (broadcast), amax = -inf, denom = 0.
// ---------------------------------------------------------------------------
__global__ void k_init(const float* __restrict__ bias,
                       float* __restrict__ out,
                       float* __restrict__ amax,
                       float* __restrict__ denom) {
    int i = blockIdx.x * blockDim.x + threadIdx.x;
    if (i < N_NODES * HC) out[i] = bias[i & (HC - 1)];
    if (i < N_NODES * HEADS) {
        amax[i]  = __int_as_float(0xff800000);  // -inf
        denom[i] = 0.f;
    }
}

// ---------------------------------------------------------------------------
// Kernel 3: per-node attention scalars  q_s/k_s[n,h] = x[n,:] @ W*_s[:,h]
// ---------------------------------------------------------------------------
__global__ void k_node_qk(const float* __restrict__ x,
                          const float* __restrict__ Wq_s,
                          const float* __restrict__ Wk_s,
                          float* __restrict__ q_s,
                          float* __restrict__ k_s) {
    int n = blockIdx.x * blockDim.x + threadIdx.x;
    if (n >= N_NODES) return;
    float q[HEADS] = {0.f, 0.f, 0.f, 0.f};
    float k[HEADS] = {0.f, 0.f, 0.f, 0.f};
    const float4* xr = (const float4*)(x + (size_t)n * DIM_IN);
    #pragma unroll 4
    for (int j4 = 0; j4 < DIM_IN / 4; ++j4) {
        float4 xv = xr[j4];
        const float* wq = Wq_s + j4 * 16;
        const float* wk = Wk_s + j4 * 16;
        #pragma unroll
        for (int h = 0; h < HEADS; ++h) {
            q[h] += xv.x * wq[h] + xv.y * wq[4 + h] + xv.z * wq[8 + h] + xv.w * wq[12 + h];
            k[h] += xv.x * wk[h] + xv.y * wk[4 + h] + xv.z * wk[8 + h] + xv.w * wk[12 + h];
        }
    }
    #pragma unroll
    for (int h = 0; h < HEADS; ++h) {
        q_s[n * HEADS + h] = q[h];
        k_s[n * HEADS + h] = k[h];
    }
}

// ---------------------------------------------------------------------------
// Kernel 4: v = x @ Wv via V_WMMA_F32_16X16X4_F32 (wave32, fp32 WMMA).
// Block = 8 waves = one 16-row x 128-col output panel.
// The 16x128 A panel (8 KB) is staged in LDS once per block (coalesced
// float4 cooperative load), then all 8 waves read their A fragments from
// LDS (ds_load) -- removes the 8x redundant global reads of x.
//
// A (16x4 f32) lane layout per ISA 7.12.2: lanes 0-15 hold M=lane,
//   VGPR0 = K=koff, VGPR1 = K=koff+1 with koff = (lane>=16) ? 2 : 0.
// B (4x16) uses the mirrored layout: lane n = lane&15, same K split.
// N_NODES % 16 == 0 and blocks are full waves, so EXEC is all ones at
// every WMMA issue.
// ---------------------------------------------------------------------------
__global__ __launch_bounds__(256)
void k_v_wmma(const float* __restrict__ x,
              const float* __restrict__ Wv,
              float* __restrict__ v) {
    __shared__ float As[16 * DIM_IN];            // 8 KB A panel
    const int tid   = threadIdx.y * 32 + threadIdx.x;   // 0..255
    const int tileM = blockIdx.x * 16;

    // Cooperative coalesced stage: 2048 floats = 512 float4 = 256 thr x 2
    {
        const float4* srcp = (const float4*)(x + (size_t)tileM * DIM_IN);
        float4* dstp = (float4*)As;
        dstp[tid]       = srcp[tid];
        dstp[tid + 256] = srcp[tid + 256];
    }
    __syncthreads();

    const int lane  = threadIdx.x;        // 0..31
    const int tileN = threadIdx.y * 16;   // 8 waves -> 8 column tiles
    const int m     = lane & 15;
    const int hi    = lane >> 4;          // 0 or 1
    const int koff  = hi * 2;

    const float* arow = As + m * DIM_IN + koff;               // A frag from LDS
    const float* bcol = Wv + (size_t)koff * HC + tileN + m;   // B frag from global

    v8f acc = {};
    #pragma unroll 8
    for (int kk = 0; kk < DIM_IN / 4; ++kk) {
        v2f a, b;
        a.x = arow[kk * 4 + 0];
        a.y = arow[kk * 4 + 1];
        b.x = bcol[(kk * 4 + 0) * HC];
        b.y = bcol[(kk * 4 + 1) * HC];
        acc = __builtin_amdgcn_wmma_f32_16x16x4_f32(
            /*neg_a=*/false, a, /*neg_b=*/false, b,
            /*c_mod=*/(short)0, acc, /*reuse_a=*/false, /*reuse_b=*/false);
    }

    // D layout: VGPR r -> (M = r + 8*hi, N = lane&15)
    #pragma unroll
    for (int r = 0; r < 8; ++r) {
        int row = tileM + r + hi * 8;
        v[(size_t)row * HC + tileN + m] = acc[r];
    }
}

// ---------------------------------------------------------------------------
// Kernel 5: per-edge logits. e_s = edge_attr[e] @ We_s, then
//   alpha = (q_s[dst]+k_s[src]+e_s)/sqrt(C) + gate*w, leaky_relu(0.2),
//   store to scratch, atomic float-max into amax[dst].
// edge_attr is the dominant DRAM stream (410 MB) -> speculative
// global_prefetch a few thousand edges ahead to keep GL2 primed.
// ---------------------------------------------------------------------------
__global__ void k_edge_alpha(const int* __restrict__ ei,
                             const float* __restrict__ ew,
                             const float* __restrict__ ea,
                             const float* __restrict__ We_s,
                             const float* __restrict__ q_s,
                             const float* __restrict__ k_s,
                             const float* __restrict__ gate,
                             float* __restrict__ alphaS,
                             float* __restrict__ amax) {
    int e = blockIdx.x * blockDim.x + threadIdx.x;
    if (e >= N_EDGES) return;

    // Prefetch the edge_attr row this thread will read ~8 blocks from now.
    int ep = e + 2048;
    if (ep < N_EDGES)
        __builtin_prefetch(ea + (size_t)ep * EDIM, 0, 1);  // global_prefetch_b8

    int src = ei[e];
    int dst = ei[N_EDGES + e];

    float es[HEADS] = {0.f, 0.f, 0.f, 0.f};
    const float4* ar = (const float4*)(ea + (size_t)e * EDIM);
    #pragma unroll 4
    for (int j4 = 0; j4 < EDIM / 4; ++j4) {
        float4 av = ar[j4];
        const float* w = We_s + j4 * 16;
        #pragma unroll
        for (int h = 0; h < HEADS; ++h)
            es[h] += av.x * w[h] + av.y * w[4 + h] + av.z * w[8 + h] + av.w * w[12 + h];
    }

    const float inv_sqrt_c = 0.17677669529663687f;  // 1/sqrt(32)
    float wgt = ew[e];
    #pragma unroll
    for (int h = 0; h < HEADS; ++h) {
        float al = (q_s[dst * HEADS + h] + k_s[src * HEADS + h] + es[h]) * inv_sqrt_c
                 + gate[h] * wgt;
        al = (al >= 0.f) ? al : 0.2f * al;          // leaky_relu(0.2)
        alphaS[(size_t)e * HEADS + h] = al;
        atomicMaxF(&amax[dst * HEADS + h], al);
    }
}

// ---------------------------------------------------------------------------
// Kernel 6: ex = exp(alpha - amax[dst]); scratch <- ex; denom[dst] += ex.
// One thread per (edge, head) -> fully coalesced scratch traffic.
// ---------------------------------------------------------------------------
__global__ void k_edge_exp(const int* __restrict__ ei,
                           float* __restrict__ alphaS,
                           const float* __restrict__ amax,
                           float* __restrict__ denom) {
    int i = blockIdx.x * blockDim.x + threadIdx.x;
    if (i >= N_EDGES * HEADS) return;
    int e = i >> 2, h = i & 3;
    int dst = ei[N_EDGES + e];
    float ex = __expf(alphaS[i] - amax[dst * HEADS + h]);
    alphaS[i] = ex;
    atomicAdd(&denom[dst * HEADS + h], ex);
}

// ---------------------------------------------------------------------------
// Kernel 7: message scatter. Wave per edge, lane per channel (4 cols/lane,
// one per head) -> 512B coalesced gather of v[src] (L2-resident: v is 51MB
// vs 192MB L2) and global_atomic_add_f32 scatter into out[dst].
// ---------------------------------------------------------------------------
__global__ __launch_bounds__(256)
void k_edge_scatter(const int* __restrict__ ei,
                    const float* __restrict__ alphaS,
                    const float* __restrict__ denom,
                    const float* __restrict__ v,
                    float* __restrict__ out) {
    int e = blockIdx.x * 8 + threadIdx.y;   // 8 waves per block
    if (e >= N_EDGES) return;
    int lane = threadIdx.x;
    int src = ei[e];
    int dst = ei[N_EDGES + e];
    #pragma unroll
    for (int h = 0; h < HEADS; ++h) {
        int col = h * 32 + lane;            // head = col/32
        float attn = alphaS[(size_t)e * HEADS + h]
                   / (denom[dst * HEADS + h] + 1e-16f);
        float msg = v[(size_t)src * HC + col] * attn;
        atomicAdd(&out[(size_t)dst * HC + col], msg);
    }
}

// ---------------------------------------------------------------------------
// Host-side launcher
// ---------------------------------------------------------------------------
extern "C" void kernel_launch(void* const* d_in, const int* in_sizes, int n_in,
                              void* d_out, int out_size, void* d_ws, size_t ws_size,
                              hipStream_t stream) {
    const float* x    = (const float*)d_in[0];
    const int*   ei   = (const int*)  d_in[1];
    const float* ew   = (const float*)d_in[2];
    const float* ea   = (const float*)d_in[3];
    const float* Wq   = (const float*)d_in[4];
    const float* Wk   = (const float*)d_in[5];
    const float* Wv   = (const float*)d_in[6];
    const float* We   = (const float*)d_in[7];
    const float* gate = (const float*)d_in[8];
    const float* bias = (const float*)d_in[9];
    float* out = (float*)d_out;

    // Workspace carve-up (~83 MB, all regenerated every call)
    char* ws = (char*)d_ws;
    float* v      = (float*)ws; ws += (size_t)N_NODES * HC    * sizeof(float);
    float* alphaS = (float*)ws; ws += (size_t)N_EDGES * HEADS * sizeof(float);
    float* q_s    = (float*)ws; ws += (size_t)N_NODES * HEADS * sizeof(float);
    float* k_s    = (float*)ws; ws += (size_t)N_NODES * HEADS * sizeof(float);
    float* amax   = (float*)ws; ws += (size_t)N_NODES * HEADS * sizeof(float);
    float* denom  = (float*)ws; ws += (size_t)N_NODES * HEADS * sizeof(float);
    float* Wq_s   = (float*)ws; ws += 512 * sizeof(float);
    float* Wk_s   = (float*)ws; ws += 512 * sizeof(float);
    float* We_s   = (float*)ws; ws += 256 * sizeof(float);

    k_reduce_w<<<5, 256, 0, stream>>>(Wq, Wk, We, Wq_s, Wk_s, We_s);
    k_init<<<(N_NODES * HC + 255) / 256, 256, 0, stream>>>(bias, out, amax, denom);
    k_node_qk<<<(N_NODES + 255) / 256, 256, 0, stream>>>(x, Wq_s, Wk_s, q_s, k_s);
    k_v_wmma<<<N_NODES / 16, dim3(32, 8), 0, stream>>>(x, Wv, v);
    k_edge_alpha<<<N_EDGES / 256, 256, 0, stream>>>(ei, ew, ea, We_s, q_s, k_s,
                                                    gate, alphaS, amax);
    k_edge_exp<<<(N_EDGES * HEADS) / 256, 256, 0, stream>>>(ei, alphaS, amax, denom);
    k_edge_scatter<<<N_EDGES / 8, dim3(32, 8), 0, stream>>>(ei, alphaS, denom, v, out);
}